// HamiltonianMapper_33131377721531
// MI455X (gfx1250) — compile-verified
//
#include <hip/hip_runtime.h>

// MI455X / gfx1250, wave32. Factored design:
//  stage 1: memory-bound streaming dot (2.56 GB read -> ~116us roofline)
//  stage 2: Z(16x25pad28) @ cgcT(25pad28 x 81pad96) via v_wmma_f32_16x16x4_f32

typedef __attribute__((ext_vector_type(2))) float v2f;
typedef __attribute__((ext_vector_type(8))) float v8f;

#define NPAIRS 200000
#define NLM    25
#define FIN    64
#define NCOL   81      // 9*9
#define ZSTR   29      // padded lm stride (odd -> LDS-bank friendly)
#define WAVES_PER_BLOCK 4

__global__ __launch_bounds__(128)
void hammap_kernel(const float* __restrict__ irr,   // (200000,2,25,64)
                   const float* __restrict__ W,     // (2,5,64,1)
                   const float* __restrict__ bptr,  // (1,)
                   const float* __restrict__ cgc,   // (9,9,25)
                   float* __restrict__ out)         // (200000,9,9)
{
  __shared__ float zbuf[WAVES_PER_BLOCK * 32 * ZSTR]; // [wave][pair*2+p][ZSTR]

  const int tid  = threadIdx.x;
  const int wave = tid >> 5;
  const int lane = tid & 31;
  const int tile = blockIdx.x * WAVES_PER_BLOCK + wave;   // 16 pairs per tile
  const int pr   = lane >> 1;       // pair-in-tile 0..15
  const int p    = lane & 1;        // p index 0..1
  const long pairIdx = (long)tile * 16 + pr;

  // ---------------- stage 1: y0[pair,p,lm] = dot64(irreps_row, W[p,deg(lm)]) --
  const float4* rowBase = (const float4*)(irr + (size_t)(pairIdx * 2 + p) * (NLM * FIN));
  const float4* Wp      = (const float4*)(W + p * 5 * FIN);

  float acc[NLM];
#pragma unroll
  for (int lm = 0; lm < NLM; ++lm) acc[lm] = 0.0f;

#pragma unroll
  for (int c = 0; c < 16; ++c) {                 // 16 float4 chunks of f
    float4 w0 = Wp[0 * 16 + c];
    float4 w1 = Wp[1 * 16 + c];
    float4 w2 = Wp[2 * 16 + c];
    float4 w3 = Wp[3 * 16 + c];
    float4 w4 = Wp[4 * 16 + c];
#pragma unroll
    for (int lm = 0; lm < NLM; ++lm) {
      const int l = (lm < 1) ? 0 : (lm < 4) ? 1 : (lm < 9) ? 2 : (lm < 16) ? 3 : 4;
      float4 w = (l == 0) ? w0 : (l == 1) ? w1 : (l == 2) ? w2 : (l == 3) ? w3 : w4;
      float4 a = rowBase[lm * 16 + c];
      acc[lm] = fmaf(a.x, w.x, acc[lm]);
      acc[lm] = fmaf(a.y, w.y, acc[lm]);
      acc[lm] = fmaf(a.z, w.z, acc[lm]);
      acc[lm] = fmaf(a.w, w.w, acc[lm]);
    }
  }
  if (p == 0) acc[0] += bptr[0];   // bias added once (p=0, lm=0)

  float* zrow = &zbuf[(wave * 32 + (pr * 2 + p)) * ZSTR];
#pragma unroll
  for (int lm = 0; lm < NLM; ++lm) zrow[lm] = acc[lm];
  zrow[25] = 0.0f; zrow[26] = 0.0f; zrow[27] = 0.0f; zrow[28] = 0.0f; // K padding

  __syncthreads();

  // ---------------- stage 2: out_tile(16x81) = Z(16x28) @ B(28x96) via WMMA ---
  // A 16x4 f32 layout: lane M = lane&15; v2f = {K=k0, K=k0+1}, k0 = ks*4 + 2*(lane>>4)
  const int m     = lane & 15;
  const int khalf = (lane >> 4) * 2;
  const float* zw = &zbuf[wave * 32 * ZSTR];

  v2f afrag[7];
#pragma unroll
  for (int ks = 0; ks < 7; ++ks) {
    const int k0 = ks * 4 + khalf;
    afrag[ks].x = zw[(m * 2 + 0) * ZSTR + k0]     + zw[(m * 2 + 1) * ZSTR + k0];     // sum over p
    afrag[ks].y = zw[(m * 2 + 0) * ZSTR + k0 + 1] + zw[(m * 2 + 1) * ZSTR + k0 + 1];
  }

  float* outRow = out + (size_t)tile * 16 * NCOL;
  const int mbase = (lane < 16) ? 0 : 8;

#pragma unroll
  for (int nt = 0; nt < 6; ++nt) {               // 81 cols -> 6 N-tiles of 16
    const int col = nt * 16 + m;
    const bool cok = (col < NCOL);
    v8f c8 = {};
#pragma unroll
    for (int ks = 0; ks < 7; ++ks) {
      const int k0 = ks * 4 + khalf;
      v2f bfrag;
      // B[k][col] = cgc[col*25 + k] (cgc fits in L0; guards give zero-padding)
      bfrag.x = (cok && (k0     < NLM)) ? cgc[col * NLM + k0]     : 0.0f;
      bfrag.y = (cok && (k0 + 1 < NLM)) ? cgc[col * NLM + k0 + 1] : 0.0f;
      c8 = __builtin_amdgcn_wmma_f32_16x16x4_f32(
               /*neg_a=*/false, afrag[ks], /*neg_b=*/false, bfrag,
               /*c_mod=*/(short)0, c8, /*reuse_a=*/false, /*reuse_b=*/false);
    }
    if (cok) {
      // D layout: lane holds N = lane&15; VGPR r holds M = r + mbase
#pragma unroll
      for (int r = 0; r < 8; ++r)
        outRow[(size_t)(r + mbase) * NCOL + col] = c8[r];
    }
  }
}

extern "C" void kernel_launch(void* const* d_in, const int* in_sizes, int n_in,
                              void* d_out, int out_size, void* d_ws, size_t ws_size,
                              hipStream_t stream) {
  const float* irr  = (const float*)d_in[0];  // neighbour_irreps
  const float* W    = (const float*)d_in[1];
  const float* b    = (const float*)d_in[2];
  const float* cgc  = (const float*)d_in[3];
  float* out        = (float*)d_out;

  // 200000 pairs = 3125 blocks * 4 waves * 16 pairs  (exact, no tail)
  hammap_kernel<<<dim3(3125), dim3(128), 0, stream>>>(irr, W, b, cgc, out);
}